// SingleT2FLS_Mamdani_11622181503714
// MI455X (gfx1250) — compile-verified
//
#include <hip/hip_runtime.h>
#include <hip/hip_bf16.h>

// ---------------------------------------------------------------------------
// T2FLS Mamdani: firing-level products rewritten as exp(quadratic form),
// quadratic form evaluated with V_WMMA_F32_16X16X4_F32 (16 samples x 16 rules,
// K = 20 padded features [x^2, x, 1]), followed by per-lane Karnik-Mendel scan.
// ---------------------------------------------------------------------------

typedef float v2f __attribute__((ext_vector_type(2)));
typedef float v8f __attribute__((ext_vector_type(8)));

#define R_RULES 16
#define A_DIMS  8
#define KSTEPS  5            // 20 padded features / 4 per WMMA step

// Coefficient matrices in B-operand layout: [kk][vgpr][lane]
//   lane 0-15 : (K = kk*4 + vgpr,     N = lane)
//   lane 16-31: (K = kk*4 + vgpr + 2, N = lane-16)
__device__ float g_BU[KSTEPS * 64];   // p_small -> UU exponent coefficients
__device__ float g_BL[KSTEPS * 64];   // p_big   -> LL exponent coefficients
__device__ float g_c1s[R_RULES];
__device__ float g_c2s[R_RULES];
__device__ int   g_idx1[R_RULES];
__device__ int   g_idx2[R_RULES];

__device__ inline float pval(const float* W, int r, int a, int mat) {
    int off = A_DIMS * r + a;
    float s1 = W[off + 1];
    float s2 = W[off + 2];
    float p1 = 1.0f / (s1 * s1);
    float p2 = 1.0f / (s2 * s2);
    // mat==0 -> UU uses min(d^2) -> p_small ; mat==1 -> LL uses p_big
    return (mat == 0) ? fminf(p1, p2) : fmaxf(p1, p2);
}

__global__ void t2fls_precompute(const float* __restrict__ W,
                                 const float* __restrict__ c1,
                                 const float* __restrict__ c2) {
    int t = threadIdx.x;
    // 2 matrices * 5 ksteps * 2 vgprs * 32 lanes = 640 coefficient slots
    for (int e = t; e < 2 * KSTEPS * 64; e += blockDim.x) {
        int mat  = e / (KSTEPS * 64);
        int rem  = e % (KSTEPS * 64);
        int kk   = rem / 64;
        int rem2 = rem % 64;
        int v    = rem2 / 32;
        int lane = rem2 % 32;
        int r    = lane & 15;
        int f    = kk * 4 + v + ((lane >> 4) ? 2 : 0);   // feature index (K)
        float coef = 0.0f;
        if (f < 8) {                       // x_f^2 term
            float p = pval(W, r, f, mat);
            coef = -0.5f * p;
        } else if (f < 16) {               // x_(f-8) term
            int a = f - 8;
            float p = pval(W, r, a, mat);
            float m = W[A_DIMS * r + a];
            coef = p * m;
        } else if (f == 16) {              // constant term
            float s = 0.0f;
            for (int a = 0; a < A_DIMS; ++a) {
                float p = pval(W, r, a, mat);
                float m = W[A_DIMS * r + a];
                s += p * m * m;
            }
            coef = -0.5f * s;
        }
        float* dst = (mat == 0) ? g_BU : g_BL;
        dst[kk * 64 + v * 32 + lane] = coef;
    }
    // stable rank-sort of c1 / c2 (matches jnp.argsort tie-breaking)
    if (t < R_RULES) {
        float v1 = c1[t];
        int rank1 = 0;
        for (int j = 0; j < R_RULES; ++j) {
            float u = c1[j];
            rank1 += (u < v1) || (u == v1 && j < t);
        }
        g_c1s[rank1] = v1;
        g_idx1[rank1] = t;

        float v2 = c2[t];
        int rank2 = 0;
        for (int j = 0; j < R_RULES; ++j) {
            float u = c2[j];
            rank2 += (u < v2) || (u == v2 && j < t);
        }
        g_c2s[rank2] = v2;
        g_idx2[rank2] = t;
    }
}

__launch_bounds__(256)
__global__ void t2fls_main(const float* __restrict__ x,
                           float* __restrict__ out,
                           int n, int ntiles) {
    __shared__ float ldsU[8][R_RULES * 16];   // [wave][sample*16 + rule]
    __shared__ float ldsL[8][R_RULES * 16];

    const int lane = threadIdx.x & 31;
    const int wv   = threadIdx.x >> 5;
    const int half = lane >> 4;       // 0: K pair {0,1} / "left";  1: {2,3} / "right"
    const int sl   = lane & 15;       // sample-in-tile (A row) and rule column (D col)

    const int waveId = blockIdx.x * (blockDim.x >> 5) + wv;
    const int nwaves = gridDim.x * (blockDim.x >> 5);

    // Hoist B operands (invariant across tiles) into registers.
    v2f bU[KSTEPS], bL[KSTEPS];
#pragma unroll
    for (int kk = 0; kk < KSTEPS; ++kk) {
        bU[kk].x = g_BU[kk * 64 + lane];
        bU[kk].y = g_BU[kk * 64 + 32 + lane];
        bL[kk].x = g_BL[kk * 64 + lane];
        bL[kk].y = g_BL[kk * 64 + 32 + lane];
    }
    // Per-half sorted centroids + permutation (left half: c1, right half: c2).
    float carr[R_RULES];
    int   iarr[R_RULES];
#pragma unroll
    for (int k = 0; k < R_RULES; ++k) {
        carr[k] = half ? g_c2s[k] : g_c1s[k];
        iarr[k] = half ? g_idx2[k] : g_idx1[k];
    }

    for (int tile = waveId; tile < ntiles; tile += nwaves) {
        int s = tile * 16 + sl;
        int sclamp = (s < n) ? s : (n - 1);
        const float4* xp = (const float4*)(x + (size_t)sclamp * A_DIMS);
        float4 xa = xp[0];
        float4 xb = xp[1];

        float f[20];
        f[0] = xa.x * xa.x;  f[1] = xa.y * xa.y;  f[2] = xa.z * xa.z;  f[3] = xa.w * xa.w;
        f[4] = xb.x * xb.x;  f[5] = xb.y * xb.y;  f[6] = xb.z * xb.z;  f[7] = xb.w * xb.w;
        f[8] = xa.x;  f[9]  = xa.y;  f[10] = xa.z;  f[11] = xa.w;
        f[12] = xb.x; f[13] = xb.y;  f[14] = xb.z;  f[15] = xb.w;
        f[16] = 1.0f; f[17] = 0.0f;  f[18] = 0.0f;  f[19] = 0.0f;

        v8f accU = {};   // log(UU) per (sample, rule)
        v8f accL = {};   // log(LL)
#pragma unroll
        for (int kk = 0; kk < KSTEPS; ++kk) {
            v2f a;
            a.x = half ? f[kk * 4 + 2] : f[kk * 4 + 0];
            a.y = half ? f[kk * 4 + 3] : f[kk * 4 + 1];
            accU = __builtin_amdgcn_wmma_f32_16x16x4_f32(
                false, a, false, bU[kk], (short)0, accU, false, false);
            accL = __builtin_amdgcn_wmma_f32_16x16x4_f32(
                false, a, false, bL[kk], (short)0, accL, false, false);
        }

        // D layout: VGPR i -> (M = i + 8*half, N = sl). exp() then scatter to LDS.
#pragma unroll
        for (int i = 0; i < 8; ++i) {
            int row = i + half * 8;                   // sample-in-tile
            ldsU[wv][row * 16 + sl] = __expf(accU[i]);
            ldsL[wv][row * 16 + sl] = __expf(accL[i]);
        }
        asm volatile("s_wait_dscnt 0x0" ::: "memory");

        // Karnik-Mendel scan: lanes 0-15 -> left endpoint of sample sl,
        // lanes 16-31 -> right endpoint of sample sl.
        float s0 = 0.0f, t0 = 0.0f;
#pragma unroll
        for (int k = 0; k < R_RULES; ++k) {
            int r = iarr[k];
            float Uv = ldsU[wv][sl * 16 + r];
            float Lv = ldsL[wv][sl * 16 + r];
            float base = half ? Uv : Lv;
            s0 += carr[k] * base;
            t0 += base;
        }
        float best = s0 / t0;
        float cs = 0.0f, ct = 0.0f;
#pragma unroll
        for (int k = 0; k < R_RULES; ++k) {
            int r = iarr[k];
            float Uv = ldsU[wv][sl * 16 + r];
            float Lv = ldsL[wv][sl * 16 + r];
            float d = half ? (Lv - Uv) : (Uv - Lv);
            cs += carr[k] * d;
            ct += d;
            float rat = (s0 + cs) / (t0 + ct);
            best = half ? fmaxf(best, rat) : fminf(best, rat);
        }

        float other = __shfl_xor(best, 16, 32);
        if (!half && s < n) {
            out[s] = 0.5f * (best + other);
        }
    }
}

extern "C" void kernel_launch(void* const* d_in, const int* in_sizes, int n_in,
                              void* d_out, int out_size, void* d_ws, size_t ws_size,
                              hipStream_t stream) {
    (void)n_in; (void)d_ws; (void)ws_size; (void)out_size;
    const float* x  = (const float*)d_in[0];   // (N, 8) float32
    const float* W  = (const float*)d_in[1];   // (384,) float32
    const float* c1 = (const float*)d_in[2];   // (16,)
    const float* c2 = (const float*)d_in[3];   // (16,)
    float* out = (float*)d_out;                // (N,) float32

    int n = in_sizes[0] / A_DIMS;
    int ntiles = (n + 15) / 16;

    t2fls_precompute<<<1, 256, 0, stream>>>(W, c1, c2);

    // 8 waves/block, aim ~4 tiles per wave
    int blocks = (ntiles + 31) / 32;
    if (blocks < 1) blocks = 1;
    if (blocks > 2048) blocks = 2048;
    t2fls_main<<<blocks, 256, 0, stream>>>(x, out, n, ntiles);
}